// HomoMessagePassing_24232205484250
// MI455X (gfx1250) — compile-verified
//
#include <hip/hip_runtime.h>
#include <stdint.h>

typedef float v4f __attribute__((ext_vector_type(4)));
// Matches the builtin's diagnosed parameter pointee type exactly:
typedef int i32x4 __attribute__((__vector_size__(16)));
typedef __attribute__((address_space(1))) i32x4* gbl_i32x4_p;   // __device__ *
typedef __attribute__((address_space(3))) i32x4* lds_i32x4_p;   // __shared__ *

#define D_FEAT 128
#define WAVES_PER_BLOCK 8          // 256 threads, wave32 -> 8 edges per block

// gfx12+ CPol field: TH[2:0], SCOPE[4:3]. TH_NT = 1 -> non-temporal at all
// cache levels: edge_attr is streamed exactly once (307 MB) and must not
// evict the L2-resident x (51 MB) / out (51 MB) working set on the 192MB L2.
#define CPOL_NT 1

// --- feature detection -------------------------------------------------------
#if defined(__has_builtin)
#if __has_builtin(__builtin_amdgcn_global_load_async_to_lds_b128)
#define HAVE_ASYNC_LDS 1
#endif
#if __has_builtin(__builtin_amdgcn_global_atomic_fadd_f32)
#define HAVE_GLOBAL_FADD 1
#endif
#endif

__device__ __forceinline__ void atomic_fadd_agent(float* p, float v) {
#if defined(HAVE_GLOBAL_FADD)
  __builtin_amdgcn_global_atomic_fadd_f32(
      (__attribute__((address_space(1))) float*)p, v);
#else
  __hip_atomic_fetch_add(p, v, __ATOMIC_RELAXED, __HIP_MEMORY_SCOPE_AGENT);
#endif
}

// ---------------------------------------------------------------------------
// Kernel 1: out = x  (vectorized b128 copy; leaves `out` resident in L2 so the
// subsequent atomic storm hits L2, not HBM)
// ---------------------------------------------------------------------------
__global__ void copy_x_kernel(const v4f* __restrict__ x,
                              v4f* __restrict__ out, long long n4) {
  long long i = (long long)blockIdx.x * blockDim.x + threadIdx.x;
  if (i < n4) out[i] = x[i];
}

// ---------------------------------------------------------------------------
// Kernel 2: one wave32 per edge.
//   lane l owns feature slice [4l, 4l+4): one b128 gather of x[src], one b128
//   of edge_attr (async-staged through LDS, non-temporal), and 4
//   global_atomic_add_f32 into out[dst].
// ---------------------------------------------------------------------------
__global__ __launch_bounds__(32 * WAVES_PER_BLOCK)
void edge_scatter_kernel(const float* __restrict__ x,
                         const float* __restrict__ edge_attr,
                         const int* __restrict__ edge_index,
                         float* __restrict__ out, int E) {
  const int lane = threadIdx.x & 31;
  const int wave = threadIdx.x >> 5;
  const long long e = (long long)blockIdx.x * WAVES_PER_BLOCK + wave;
  const bool live = (e < (long long)E);

#if defined(HAVE_ASYNC_LDS)
  // Stage this wave's edge_attr row (512B) into LDS asynchronously (ASYNCcnt).
  // Each wave stages only its own row -> no cross-wave dependency, no barrier;
  // per-wave s_wait_asynccnt suffices. Issued BEFORE the index loads + random
  // x[src] gather so the stream overlaps the high-latency gather.
  __shared__ v4f s_ea[WAVES_PER_BLOCK * 32];
  if (live) {
    const float* g = edge_attr + e * D_FEAT + (lane << 2);
    __builtin_amdgcn_global_load_async_to_lds_b128(
        (gbl_i32x4_p)g,
        (lds_i32x4_p)&s_ea[(wave << 5) + lane],
        /*imm_offset=*/0, /*cpol=*/CPOL_NT);
  }
#endif

  if (!live) return;

  // Per-wave-uniform indices (all lanes load the same dwords; L0 broadcasts).
  const int src = edge_index[e];
  const int dst = edge_index[(long long)E + e];

  // Random gather of x[src]: regular (temporal) load -> wants to live in L2.
  const v4f xs = *(const v4f*)(x + (long long)src * D_FEAT + (lane << 2));

  v4f ea;
#if defined(HAVE_ASYNC_LDS)
  asm volatile("s_wait_asynccnt 0" ::: "memory");
  ea = s_ea[(wave << 5) + lane];
#else
  // Streamed exactly once: non-temporal so it doesn't evict x/out from L2.
  ea = __builtin_nontemporal_load(
      (const v4f*)(edge_attr + e * D_FEAT + (lane << 2)));
#endif

  const v4f m = xs + ea;

  float* o = out + (long long)dst * D_FEAT + (lane << 2);
  atomic_fadd_agent(o + 0, m.x);
  atomic_fadd_agent(o + 1, m.y);
  atomic_fadd_agent(o + 2, m.z);
  atomic_fadd_agent(o + 3, m.w);
}

// ---------------------------------------------------------------------------
extern "C" void kernel_launch(void* const* d_in, const int* in_sizes, int n_in,
                              void* d_out, int out_size, void* d_ws,
                              size_t ws_size, hipStream_t stream) {
  const float* x         = (const float*)d_in[0];   // [N, 128] f32
  const float* edge_attr = (const float*)d_in[1];   // [E, 128] f32
  const int*   edge_idx  = (const int*)d_in[2];     // [2, E] int32
  float*       out       = (float*)d_out;           // [N, 128] f32

  const long long nd = (long long)in_sizes[0];      // N * 128
  const int       E  = in_sizes[2] / 2;

  // Kernel 1: out = x
  const long long n4 = nd / 4;
  const int tb = 256;
  const int cb = (int)((n4 + tb - 1) / tb);
  copy_x_kernel<<<cb, tb, 0, stream>>>((const v4f*)x, (v4f*)out, n4);

  // Kernel 2: scatter messages
  const int eb = (E + WAVES_PER_BLOCK - 1) / WAVES_PER_BLOCK;
  edge_scatter_kernel<<<eb, 32 * WAVES_PER_BLOCK, 0, stream>>>(
      x, edge_attr, edge_idx, out, E);
}